// GIN_27977416966474
// MI455X (gfx1250) — compile-verified
//
#include <hip/hip_runtime.h>
#include <hip/hip_bf16.h>

typedef float v2f __attribute__((ext_vector_type(2)));
typedef float v8f __attribute__((ext_vector_type(8)));

#define N_NODES   100000
#define N_EDGES   1600000
#define N_GRAPHS  512
#define DIM       128
#define OUT_DIM   10
#define BN_EPS    1e-5f

// ---------------------------------------------------------------------------
// zero fill
__global__ void k_zero(float* __restrict__ p, int n) {
    int i = blockIdx.x * blockDim.x + threadIdx.x;
    if (i < n) p[i] = 0.0f;
}

// copy (float4) : AGG = X
__global__ void k_copy4(const float* __restrict__ src, float* __restrict__ dst, int n4) {
    int i = blockIdx.x * blockDim.x + threadIdx.x;
    if (i < n4) ((float4*)dst)[i] = ((const float4*)src)[i];
}

// ---------------------------------------------------------------------------
// Edge scatter: one wave32 per edge, lane l handles features [4l, 4l+3].
// AGG[dst] += X[src]  (AGG pre-initialized with X, giving h = x + sum_j x_j)
__global__ void k_edge_scatter(const float* __restrict__ X,
                               const int* __restrict__ src,
                               const int* __restrict__ dst,
                               float* __restrict__ AGG, int nE) {
    int e = blockIdx.x * 8 + (threadIdx.x >> 5);
    if (e >= nE) return;
    int lane = threadIdx.x & 31;
    int s = src[e];
    int d = dst[e];
    float4 v = ((const float4*)(X + (size_t)s * DIM))[lane];
    float* o = AGG + (size_t)d * DIM + lane * 4;
    atomicAdd(o + 0, v.x);
    atomicAdd(o + 1, v.y);
    atomicAdd(o + 2, v.z);
    atomicAdd(o + 3, v.w);
}

// ---------------------------------------------------------------------------
// WMMA f32 GEMM:  out[M x 128] = act(affine(A)) @ W[128 x 128] + bias
// Block = 256 threads (8 waves). Block tile: (NTILE*16) rows x 128 cols.
// Each wave owns one 16-col N-tile and NTILE M-tiles, so each loaded B
// fragment feeds NTILE v_wmma ops. NTILE is a compile-time template param so
// the inner loop is completely branch-free (no per-step SALU guards).
// Optional per-column input affine (BN fold) + ReLU fused into LDS staging.
#define LDS_STRIDE 132   // 128 + 4 pad -> conflict-free wmma A reads
template <int NTILE>
__global__ void k_gemm128_wmma(const float* __restrict__ A,
                               const float* __restrict__ W,     // [128][128] row-major (k,n)
                               const float* __restrict__ bias,  // [128]
                               const float* __restrict__ sc,    // nullable [128]
                               const float* __restrict__ sh,    // nullable [128]
                               int relu_in, int relu_out,
                               float* __restrict__ out, int m_base) {
    constexpr int ROWS = NTILE * 16;
    __shared__ float sA[ROWS * LDS_STRIDE];
    const int tid = threadIdx.x;
    const int m0  = m_base + blockIdx.x * ROWS;

    // stage ROWS x 128 A tile (float2 granularity), fusing BN affine + ReLU
    for (int i = tid; i < ROWS * 64; i += 256) {
        int m  = i >> 6;
        int c2 = i & 63;
        int c  = c2 * 2;
        float2 v = ((const float2*)(A + (size_t)(m0 + m) * DIM))[c2];
        if (sc) {
            v.x = v.x * sc[c]     + sh[c];
            v.y = v.y * sc[c + 1] + sh[c + 1];
        }
        if (relu_in) {
            v.x = fmaxf(v.x, 0.0f);
            v.y = fmaxf(v.y, 0.0f);
        }
        *(float2*)&sA[m * LDS_STRIDE + c] = v;
    }
    __syncthreads();

    const int lane = tid & 31;
    const int wave = tid >> 5;        // 0..7 -> N tile
    const int nb   = wave * 16;
    const int n    = lane & 15;
    const int half = lane >> 4;       // 0: K k..k+1 / M 0..7 ; 1: K k+2..k+3 / M 8..15
    const int mrow = lane & 15;       // A-fragment row within tile

    v8f acc[NTILE];
#pragma unroll
    for (int t = 0; t < NTILE; ++t) acc[t] = (v8f){};

#pragma unroll 4
    for (int k = 0; k < DIM; k += 4) {
        const int ka = k + half * 2;
        // B 4x16 f32 fragment (row k striped over lanes, half split like A)
        v2f b;
        b.x = W[(size_t)ka * DIM + nb + n];
        b.y = W[(size_t)(ka + 1) * DIM + nb + n];
#pragma unroll
        for (int t = 0; t < NTILE; ++t) {
            v2f a;
            a.x = sA[(t * 16 + mrow) * LDS_STRIDE + ka];
            a.y = sA[(t * 16 + mrow) * LDS_STRIDE + ka + 1];
            acc[t] = __builtin_amdgcn_wmma_f32_16x16x4_f32(false, a, false, b,
                                                           (short)0, acc[t], false, false);
        }
    }

    // C/D layout: vgpr r -> row (half*8 + r) within tile, col = nb + n
    const float bv = bias[nb + n];
#pragma unroll
    for (int t = 0; t < NTILE; ++t) {
        const size_t rbase = (size_t)(m0 + t * 16 + half * 8);
#pragma unroll
        for (int r = 0; r < 8; ++r) {
            float v = acc[t][r] + bv;
            if (relu_out) v = fmaxf(v, 0.0f);
            out[(rbase + r) * DIM + nb + n] = v;
        }
    }
}

// host-side dispatch: bulk with NTILE=4, 16-row-granular tail specialized
static inline void launch_gemm(const float* A, const float* W, const float* bias,
                               const float* sc, const float* sh,
                               int relu_in, int relu_out, float* out, int M,
                               hipStream_t stream) {
    const int full = M / 64;             // blocks of 64 rows
    const int rem_tiles = (M - full * 64) / 16;   // 0..3 remaining 16-row tiles
    if (full > 0)
        k_gemm128_wmma<4><<<full, 256, 0, stream>>>(A, W, bias, sc, sh,
                                                    relu_in, relu_out, out, 0);
    const int mb = full * 64;
    switch (rem_tiles) {
        case 1: k_gemm128_wmma<1><<<1, 256, 0, stream>>>(A, W, bias, sc, sh,
                                                         relu_in, relu_out, out, mb); break;
        case 2: k_gemm128_wmma<2><<<1, 256, 0, stream>>>(A, W, bias, sc, sh,
                                                         relu_in, relu_out, out, mb); break;
        case 3: k_gemm128_wmma<3><<<1, 256, 0, stream>>>(A, W, bias, sc, sh,
                                                         relu_in, relu_out, out, mb); break;
        default: break;
    }
}

// ---------------------------------------------------------------------------
// Column sum / sum-of-squares for BN (block reduces 256 rows, global atomics)
#define STAT_ROWS 256
__global__ void k_colstats(const float* __restrict__ H, int M,
                           float* __restrict__ csum, float* __restrict__ csq) {
    __shared__ float ls[DIM], lq[DIM];
    int col = threadIdx.x & 127;
    int sub = threadIdx.x >> 7;   // 0/1
    if (threadIdx.x < DIM) { ls[threadIdx.x] = 0.0f; lq[threadIdx.x] = 0.0f; }
    __syncthreads();
    int r0   = blockIdx.x * STAT_ROWS;
    int rend = r0 + STAT_ROWS; if (rend > M) rend = M;
    float s = 0.0f, q = 0.0f;
    for (int r = r0 + sub; r < rend; r += 2) {
        float v = H[(size_t)r * DIM + col];
        s += v; q += v * v;
    }
    atomicAdd(&ls[col], s);
    atomicAdd(&lq[col], q);
    __syncthreads();
    if (threadIdx.x < DIM) {
        atomicAdd(&csum[threadIdx.x], ls[threadIdx.x]);
        atomicAdd(&csq[threadIdx.x],  lq[threadIdx.x]);
    }
}

// Fold BN into per-column affine: y = h*scb[c] + shb[c]
__global__ void k_bn_final(const float* __restrict__ csum, const float* __restrict__ csq,
                           const float* __restrict__ g, const float* __restrict__ be,
                           float* __restrict__ scb, float* __restrict__ shb, int M) {
    int c = threadIdx.x;
    if (c >= DIM) return;
    float invM = 1.0f / (float)M;
    float mu   = csum[c] * invM;
    float var  = csq[c] * invM - mu * mu;
    float rstd = rsqrtf(var + BN_EPS);
    float s    = g[c] * rstd;
    scb[c] = s;
    shb[c] = be[c] - mu * s;
}

// ---------------------------------------------------------------------------
// Pooling
__global__ void k_pool_cnt(const int* __restrict__ batch, float* __restrict__ cnts, int N) {
    int n = blockIdx.x * blockDim.x + threadIdx.x;
    if (n < N) atomicAdd(&cnts[batch[n]], 1.0f);
}

__global__ void k_pool_sum(const float* __restrict__ H, const int* __restrict__ batch,
                           float* __restrict__ sums, int N) {
    int nidx = blockIdx.x * 8 + (threadIdx.x >> 5);
    if (nidx >= N) return;
    int lane = threadIdx.x & 31;
    int g = batch[nidx];
    float4 v = ((const float4*)(H + (size_t)nidx * DIM))[lane];
    float* o = sums + (size_t)g * DIM + lane * 4;
    atomicAdd(o + 0, v.x);
    atomicAdd(o + 1, v.y);
    atomicAdd(o + 2, v.z);
    atomicAdd(o + 3, v.w);
}

__global__ void k_pool_final(const float* __restrict__ sums, const float* __restrict__ cnts,
                             float* __restrict__ reps) {
    int i = blockIdx.x * blockDim.x + threadIdx.x;
    if (i < N_GRAPHS * DIM) {
        int g = i >> 7;
        reps[i] = sums[i] / fmaxf(cnts[g], 1.0f);
    }
}

// Classifier head: logits[512 x 10] = z1[512 x 128] @ cw2[128 x 10] + cb2
__global__ void k_logits(const float* __restrict__ z1, const float* __restrict__ cw2,
                         const float* __restrict__ cb2, float* __restrict__ out) {
    int i = blockIdx.x * blockDim.x + threadIdx.x;
    if (i >= N_GRAPHS * OUT_DIM) return;
    int g = i / OUT_DIM, o = i % OUT_DIM;
    float acc = cb2[o];
#pragma unroll 8
    for (int k = 0; k < DIM; ++k) acc += z1[(size_t)g * DIM + k] * cw2[k * OUT_DIM + o];
    out[i] = acc;
}

// ---------------------------------------------------------------------------
extern "C" void kernel_launch(void* const* d_in, const int* in_sizes, int n_in,
                              void* d_out, int out_size, void* d_ws, size_t ws_size,
                              hipStream_t stream) {
    const float* x     = (const float*)d_in[0];
    const int*   ei    = (const int*)d_in[1];
    const int*   batch = (const int*)d_in[2];
    const int nE = in_sizes[1] / 2;
    const int nN = in_sizes[0] / DIM;
    const int* src = ei;
    const int* dst = ei + nE;

    const float* w1[3] = { (const float*)d_in[3],  (const float*)d_in[9],  (const float*)d_in[15] };
    const float* b1[3] = { (const float*)d_in[4],  (const float*)d_in[10], (const float*)d_in[16] };
    const float* gm[3] = { (const float*)d_in[5],  (const float*)d_in[11], (const float*)d_in[17] };
    const float* be[3] = { (const float*)d_in[6],  (const float*)d_in[12], (const float*)d_in[18] };
    const float* w2[3] = { (const float*)d_in[7],  (const float*)d_in[13], (const float*)d_in[19] };
    const float* b2[3] = { (const float*)d_in[8],  (const float*)d_in[14], (const float*)d_in[20] };
    const float* cw1 = (const float*)d_in[21];
    const float* cb1 = (const float*)d_in[22];
    const float* cw2 = (const float*)d_in[23];
    const float* cb2 = (const float*)d_in[24];

    // workspace carve-up
    const size_t NF = (size_t)nN * DIM;
    float* wsf  = (float*)d_ws;
    float* AGG  = wsf;
    float* H1   = wsf + NF;
    float* Y    = wsf + 2 * NF;
    float* csum = wsf + 3 * NF;
    float* csq  = csum + DIM;
    float* scb  = csq + DIM;
    float* shb  = scb + DIM;
    float* psum = shb + DIM;                 // [512][128]
    float* cnts = psum + N_GRAPHS * DIM;     // [512]
    float* z1   = cnts + N_GRAPHS;           // [512][128]

    // outputs
    float* h_out     = (float*)d_out;
    float* reps_out  = h_out + NF;
    float* logit_out = reps_out + N_GRAPHS * DIM;

    const int n4 = (int)(NF / 4);

    // zero pooling accumulators every call (workspace persists across replays)
    k_zero<<<(N_GRAPHS * DIM + N_GRAPHS + 255) / 256, 256, 0, stream>>>(psum, N_GRAPHS * DIM + N_GRAPHS);

    const float* inX[3]  = { x, Y, Y };
    float*       outY[3] = { Y, Y, h_out };

    for (int L = 0; L < 3; ++L) {
        // AGG = X ; AGG[dst] += X[src]
        k_copy4<<<(n4 + 255) / 256, 256, 0, stream>>>(inX[L], AGG, n4);
        k_edge_scatter<<<(nE + 7) / 8, 256, 0, stream>>>(inX[L], src, dst, AGG, nE);
        // H1 = AGG @ w1 + b1   (WMMA f32)
        launch_gemm(AGG, w1[L], b1[L], nullptr, nullptr, 0, 0, H1, nN, stream);
        // BN stats -> per-column affine
        k_zero<<<1, 256, 0, stream>>>(csum, 2 * DIM);
        k_colstats<<<(nN + STAT_ROWS - 1) / STAT_ROWS, 256, 0, stream>>>(H1, nN, csum, csq);
        k_bn_final<<<1, DIM, 0, stream>>>(csum, csq, gm[L], be[L], scb, shb, nN);
        // Y = act( relu(BN(H1)) @ w2 + b2 )   (BN+ReLU fused into A staging)
        launch_gemm(H1, w2[L], b2[L], scb, shb, 1, (L < 2) ? 1 : 0, outY[L], nN, stream);
    }

    // global mean pool
    k_pool_cnt<<<(nN + 255) / 256, 256, 0, stream>>>(batch, cnts, nN);
    k_pool_sum<<<(nN + 7) / 8, 256, 0, stream>>>(h_out, batch, psum, nN);
    k_pool_final<<<(N_GRAPHS * DIM + 255) / 256, 256, 0, stream>>>(psum, cnts, reps_out);

    // classifier: z1 = relu(reps @ cw1 + cb1)  (WMMA), logits = z1 @ cw2 + cb2
    launch_gemm(reps_out, cw1, cb1, nullptr, nullptr, 0, 1, z1, N_GRAPHS, stream);
    k_logits<<<(N_GRAPHS * OUT_DIM + 255) / 256, 256, 0, stream>>>(z1, cw2, cb2, logit_out);
}